// Optim3_72224170049946
// MI455X (gfx1250) — compile-verified
//
#include <hip/hip_runtime.h>
#include <hip/hip_bf16.h>

// ---------------------------------------------------------------------------
// Linear state-space model, fp32 WMMA (V_WMMA_F32_16X16X4_F32).
//
// Phase 1/3: parallel GEMMs (M=262144), HBM-bound (~0.5GB @ 23.3TB/s).
// Phase 2: 4096-step dependent scan; W_A resident in LDS (320KB/WGP),
// 4 WGs x 16 batches, double-buffered x, one barrier per step.
// This revision removes all branches from the K-loops (segment split at
// compile time) and shares A fragments across both N-tiles per wave.
// ---------------------------------------------------------------------------

typedef __attribute__((ext_vector_type(2))) float v2f;
typedef __attribute__((ext_vector_type(4))) float v4f;
typedef __attribute__((ext_vector_type(8))) float v8f;

#define BDIM 64
#define SDIM 4096
#define XB   64
#define XN   256
#define YDIM 128
#define UDIM 64
#define YH   128

// padded LDS row stride (floats): 260 mod 64 = 4 -> rows m=0..15 map to
// distinct banks for the strided fragment reads; 260*4B keeps 8B alignment.
#define LROW 260

static __device__ __forceinline__ v8f wmma_f32(v2f a, v2f b, v8f c) {
  // (neg_a, A, neg_b, B, c_mod, C, reuse_a, reuse_b)
  return __builtin_amdgcn_wmma_f32_16x16x4_f32(false, a, false, b, (short)0, c,
                                               false, false);
}

// ---------------------------------------------------------------------------
// Kernel 1: d[m,n] = [y|behav|u][m,:] . [W_K|W_B][n,:] + (bA+bK+bB)[n]
// Block = one 16-row M tile; wave w owns N-columns [w*32, w*32+32).
// Three branch-free K segment loops; A fragment shared by both N-tiles.
// ---------------------------------------------------------------------------
__global__ __launch_bounds__(256) void k_compute_d(
    const float* __restrict__ behav, const float* __restrict__ y,
    const float* __restrict__ u, const float* __restrict__ W_K,
    const float* __restrict__ W_B, const float* __restrict__ b_A,
    const float* __restrict__ b_K, const float* __restrict__ b_B,
    float* __restrict__ dbuf) {
  const int lane = threadIdx.x & 31;
  const int wave = threadIdx.x >> 5;
  const int g = lane >> 4;   // lane group: K-pair select / C row half
  const int lm = lane & 15;
  const int row = blockIdx.x * 16 + lm;  // A-fragment row (m = b*S+s)
  const int n0 = wave * 32 + lm;         // N column, tile 0
  const int n1 = n0 + 16;                // N column, tile 1

  const float bias0 = b_A[n0] + b_K[n0] + b_B[n0];
  const float bias1 = b_A[n1] + b_K[n1] + b_B[n1];
  v8f c0 = {bias0, bias0, bias0, bias0, bias0, bias0, bias0, bias0};
  v8f c1 = {bias1, bias1, bias1, bias1, bias1, bias1, bias1, bias1};

  // hoisted per-lane base pointers (all 8B aligned: offsets even)
  const float* ay  = y     + (size_t)row * YDIM + g * 2;
  const float* ab  = behav + (size_t)row * XB   + g * 2;
  const float* au  = u     + (size_t)row * UDIM + g * 2;
  const float* wk0 = W_K + (size_t)n0 * (YDIM + XB) + g * 2;
  const float* wk1 = W_K + (size_t)n1 * (YDIM + XB) + g * 2;
  const float* wb0 = W_B + (size_t)n0 * UDIM + g * 2;
  const float* wb1 = W_B + (size_t)n1 * UDIM + g * 2;

  // Segment 1: y, K = 0..127 (kk = 0..31)
#pragma unroll 8
  for (int kk = 0; kk < 32; ++kk) {
    v2f a = *(const v2f*)(ay + kk * 4);
    c0 = wmma_f32(a, *(const v2f*)(wk0 + kk * 4), c0);
    c1 = wmma_f32(a, *(const v2f*)(wk1 + kk * 4), c1);
  }
  // Segment 2: behav, K = 128..191 (kk = 32..47)
#pragma unroll 8
  for (int kk = 0; kk < 16; ++kk) {
    v2f a = *(const v2f*)(ab + kk * 4);
    c0 = wmma_f32(a, *(const v2f*)(wk0 + 128 + kk * 4), c0);
    c1 = wmma_f32(a, *(const v2f*)(wk1 + 128 + kk * 4), c1);
  }
  // Segment 3: u, K = 192..255 (kk = 48..63)
#pragma unroll 8
  for (int kk = 0; kk < 16; ++kk) {
    v2f a = *(const v2f*)(au + kk * 4);
    c0 = wmma_f32(a, *(const v2f*)(wb0 + kk * 4), c0);
    c1 = wmma_f32(a, *(const v2f*)(wb1 + kk * 4), c1);
  }

  const int mb = blockIdx.x * 16 + g * 8;  // C rows: m = r + g*8
#pragma unroll
  for (int r = 0; r < 8; ++r) {
    dbuf[(size_t)(mb + r) * XN + n0] = c0[r];
    dbuf[(size_t)(mb + r) * XN + n1] = c1[r];
  }
}

// ---------------------------------------------------------------------------
// Kernel 2: scan x_{t+1} = x_t W_A^T + d_t, emit x_t. Block = 16 batches,
// W_A staged in padded LDS, x double-buffered in LDS, wave owns 2 N-tiles.
// ---------------------------------------------------------------------------
__global__ __launch_bounds__(256) void k_scan(const float* __restrict__ W_A,
                                              const float* __restrict__ dbuf,
                                              float* __restrict__ xs) {
  extern __shared__ float smem[];
  float* Wa = smem;               // XN rows x LROW
  float* xb = smem + XN * LROW;   // 2 x 16 x LROW
  const int tid = threadIdx.x;
  const int lane = tid & 31, wave = tid >> 5;
  const int g = lane >> 4, lm = lane & 15;
  const int batch0 = blockIdx.x * 16;
  const int n0 = wave * 32 + lm;
  const int n1 = n0 + 16;

  // stage W_A into padded LDS rows
  for (int i = tid * 4; i < XN * XN; i += 256 * 4) {
    const int n = i >> 8, k = i & 255;
    *(v4f*)(Wa + n * LROW + k) = *(const v4f*)(W_A + i);
  }
  // x_0 = 0
  for (int i = tid; i < 16 * LROW; i += 256) xb[i] = 0.f;
  __syncthreads();

  // hoisted LDS fragment pointers
  const float* wa0 = Wa + n0 * LROW + g * 2;
  const float* wa1 = Wa + n1 * LROW + g * 2;
  const int aoff = lm * LROW + g * 2;

  v8f cp0 = (v8f){0, 0, 0, 0, 0, 0, 0, 0};
  v8f cp1 = cp0;

  int cur = 0;
  for (int t = 0; t < SDIM; ++t) {
    const float* xcur = xb + cur * 16 * LROW;
    float* xnxt = xb + (cur ^ 1) * 16 * LROW;

    // fresh accumulators = d_t tiles (both N-tiles)
    v8f c0, c1;
#pragma unroll
    for (int r = 0; r < 8; ++r) {
      const size_t base = ((size_t)(batch0 + g * 8 + r) * SDIM + t) * XN;
      c0[r] = dbuf[base + n0];
      c1[r] = dbuf[base + n1];
    }
    // d_t fully read before overwrite (xs may alias dbuf)
    asm volatile("s_wait_loadcnt 0x0" ::: "memory");
    // emit x_t
#pragma unroll
    for (int r = 0; r < 8; ++r) {
      const size_t base = ((size_t)(batch0 + g * 8 + r) * SDIM + t) * XN;
      xs[base + n0] = cp0[r];
      xs[base + n1] = cp1[r];
    }
    // x_{t+1} = x_t W_A^T + d_t ; shared A fragment, two B fragments
#pragma unroll 8
    for (int kk = 0; kk < 64; ++kk) {
      v2f a = *(const v2f*)(xcur + aoff + kk * 4);
      c0 = wmma_f32(a, *(const v2f*)(wa0 + kk * 4), c0);
      c1 = wmma_f32(a, *(const v2f*)(wa1 + kk * 4), c1);
    }
    // publish x_{t+1} slice to the other buffer
#pragma unroll
    for (int r = 0; r < 8; ++r) {
      xnxt[(g * 8 + r) * LROW + n0] = c0[r];
      xnxt[(g * 8 + r) * LROW + n1] = c1[r];
    }
    cp0 = c0;
    cp1 = c1;
    __syncthreads();
    cur ^= 1;
  }
}

// ---------------------------------------------------------------------------
// Kernel 3: yhat[m,n] = xs[m,:] . W_C[n,:] + b_C[n]; 8 waves x 16 cols = 128.
// ---------------------------------------------------------------------------
__global__ __launch_bounds__(256) void k_yhat(const float* __restrict__ xs,
                                              const float* __restrict__ W_C,
                                              const float* __restrict__ b_C,
                                              float* __restrict__ yh) {
  const int lane = threadIdx.x & 31, wave = threadIdx.x >> 5;
  const int g = lane >> 4, lm = lane & 15;
  const int row = blockIdx.x * 16 + lm;
  const int ncol = wave * 16 + lm;
  const float bias = b_C[ncol];
  v8f c = {bias, bias, bias, bias, bias, bias, bias, bias};

  const float* ap = xs + (size_t)row * XN + g * 2;
  const float* bp = W_C + (size_t)ncol * XN + g * 2;
#pragma unroll 8
  for (int kk = 0; kk < 64; ++kk) {
    v2f a = *(const v2f*)(ap + kk * 4);
    c = wmma_f32(a, *(const v2f*)(bp + kk * 4), c);
  }
  const int mb = blockIdx.x * 16 + g * 8;
#pragma unroll
  for (int r = 0; r < 8; ++r)
    yh[(size_t)(mb + r) * YH + ncol] = c[r];
}

extern "C" void kernel_launch(void* const* d_in, const int* in_sizes, int n_in,
                              void* d_out, int out_size, void* d_ws,
                              size_t ws_size, hipStream_t stream) {
  (void)in_sizes; (void)n_in; (void)out_size;
  const float* behav = (const float*)d_in[0];
  const float* y     = (const float*)d_in[1];
  const float* u     = (const float*)d_in[2];
  const float* W_A   = (const float*)d_in[3];
  const float* b_A   = (const float*)d_in[4];
  const float* W_K   = (const float*)d_in[5];
  const float* b_K   = (const float*)d_in[6];
  const float* W_B   = (const float*)d_in[7];
  const float* b_B   = (const float*)d_in[8];
  const float* W_C   = (const float*)d_in[9];
  const float* b_C   = (const float*)d_in[10];

  float* out = (float*)d_out;
  float* yh = out;                               // [B,S,YH]
  float* xs = out + (size_t)BDIM * SDIM * YH;    // [B,S,XN]

  // d buffer: 256MB. Prefer scratch; else alias the xs output region
  // (safe: scan reads d[b,t,:] before overwriting the slot with x_t).
  const size_t dBytes = (size_t)BDIM * SDIM * XN * sizeof(float);
  float* dbuf = (ws_size >= dBytes) ? (float*)d_ws : xs;

  dim3 blk(256);
  k_compute_d<<<BDIM * SDIM / 16, blk, 0, stream>>>(behav, y, u, W_K, W_B, b_A,
                                                    b_K, b_B, dbuf);
  const size_t lds = (size_t)(XN * LROW + 2 * 16 * LROW) * sizeof(float);
  k_scan<<<BDIM / 16, blk, lds, stream>>>(W_A, dbuf, xs);
  k_yhat<<<BDIM * SDIM / 16, blk, 0, stream>>>(xs, W_C, b_C, yh);
}